// TTTLinnerLayer_58480274702561
// MI455X (gfx1250) — compile-verified
//
#include <hip/hip_runtime.h>
#include <math.h>

// Problem constants (from reference): B=2, S=256, H=768, 4H=3072, chunk C=64.
#define BB 2
#define SS 256
#define HH 768
#define H4 3072
#define CHK 64
#define LRC 0.01f
#define EPSC 1e-5f

typedef float v2f __attribute__((ext_vector_type(2)));
typedef float v8f __attribute__((ext_vector_type(8)));

// ---------------- fp32 WMMA 16x16x4 helpers (gfx1250, wave32) ----------------
__device__ __forceinline__ v8f wmma4(v2f a, v2f b, v8f c) {
  // V_WMMA_F32_16X16X4_F32 : D = A(16x4) * B(4x16) + C(16x16), all f32.
  return __builtin_amdgcn_wmma_f32_16x16x4_f32(
      /*neg_a=*/false, a, /*neg_b=*/false, b,
      /*c_mod=*/(short)0, c, /*reuse_a=*/false, /*reuse_b=*/false);
}

// Operand stored as [rows x K] row-major (A row-major, or B given as N-major
// "B-transposed" weight): lane l -> element [r0 + (l&15)][k0 + 2*(l>>4) + j]
__device__ __forceinline__ v2f frag_nk(const float* M, int ld, int r0, int k0, int lane) {
  const float* p = M + (size_t)(r0 + (lane & 15)) * ld + k0 + ((lane >> 4) << 1);
  v2f v; v.x = p[0]; v.y = p[1];
  return v;
}
// Operand stored as [K x rows] (K-major): lane l -> element [r0+(l&15)] of
// k-rows k0 + 2*(l>>4) + j  (stride-ld between the two j elements)
__device__ __forceinline__ v2f frag_kn(const float* M, int ld, int k0, int r0, int lane) {
  const float* p = M + (size_t)(k0 + ((lane >> 4) << 1)) * ld + r0 + (lane & 15);
  v2f v; v.x = p[0]; v.y = p[ld];
  return v;
}
// C/D 16x16 f32 tile: VGPR i -> M = i + 8*(lane>>4), N = lane&15
__device__ __forceinline__ void store_tile(float* D, int ld, int r0, int c0, v8f v, int lane) {
  int n = c0 + (lane & 15);
  int m0 = r0 + ((lane >> 4) << 3);
#pragma unroll
  for (int i = 0; i < 8; ++i) D[(size_t)(m0 + i) * ld + n] = v[i];
}
__device__ __forceinline__ v8f load_tile(const float* D, int ld, int r0, int c0, int lane) {
  int n = c0 + (lane & 15);
  int m0 = r0 + ((lane >> 4) << 3);
  v8f v;
#pragma unroll
  for (int i = 0; i < 8; ++i) v[i] = D[(size_t)(m0 + i) * ld + n];
  return v;
}

// ---------------- kernels ----------------

// Replicate W_init/b_init into per-batch fast-weight state.
__global__ void k_init(const float* __restrict__ Wi, const float* __restrict__ bi,
                       float* __restrict__ Wst, float* __restrict__ bst) {
  int i = blockIdx.x * blockDim.x + threadIdx.x;
  const int n = HH * HH;
  if (i < n) {
    float w = Wi[i];
    Wst[i] = w;
    Wst[n + i] = w;
  }
  if (i < HH) {
    float b = bi[i];
    bst[i] = b;
    bst[HH + i] = b;
  }
}

// P0 = X_c * Wst^T + bst   [C x H] per batch.  One wave per 16x16 tile.
__global__ void k_p0(const float* __restrict__ x, const float* __restrict__ Wst,
                     const float* __restrict__ bst, float* __restrict__ P0, int chunk) {
  int wid = (blockIdx.x * blockDim.x + threadIdx.x) >> 5;
  int lane = threadIdx.x & 31;
  int tn = wid % (HH / 16);
  int tm = (wid / (HH / 16)) % (CHK / 16);
  int b  = wid / ((HH / 16) * (CHK / 16));
  const float* Xc = x + ((size_t)b * SS + chunk * CHK) * HH;
  const float* W  = Wst + (size_t)b * HH * HH;
  v8f acc = {};
  for (int k0 = 0; k0 < HH; k0 += 4) {
    v2f a  = frag_nk(Xc, HH, tm * 16, k0, lane);   // A[m,k] = Xc[m,k]
    v2f bb = frag_nk(W,  HH, tn * 16, k0, lane);   // B[k,n] = W[n,k]
    acc = wmma4(a, bb, acc);
  }
  float bias = bst[(size_t)b * HH + tn * 16 + (lane & 15)];
#pragma unroll
  for (int i = 0; i < 8; ++i) acc[i] += bias;
  store_tile(P0 + (size_t)b * CHK * HH, HH, tm * 16, tn * 16, acc, lane);
}

// A[t,s] = (s<t) ? LR*(x_s . x_t + 1) : 0   [C x C] per batch (strict lower).
__global__ void k_attn(const float* __restrict__ x, float* __restrict__ Amat, int chunk) {
  int wid = (blockIdx.x * blockDim.x + threadIdx.x) >> 5;
  int lane = threadIdx.x & 31;
  int tn = wid % (CHK / 16);
  int tm = (wid / (CHK / 16)) % (CHK / 16);
  int b  = wid / ((CHK / 16) * (CHK / 16));
  const float* Xc = x + ((size_t)b * SS + chunk * CHK) * HH;
  v8f acc = {};
  for (int k0 = 0; k0 < HH; k0 += 4) {
    v2f a  = frag_nk(Xc, HH, tm * 16, k0, lane);
    v2f bb = frag_nk(Xc, HH, tn * 16, k0, lane);   // B[k,s] = Xc[s,k]
    acc = wmma4(a, bb, acc);
  }
  int s = tn * 16 + (lane & 15);
  int mbase = tm * 16 + ((lane >> 4) << 3);
  float* Ab = Amat + (size_t)b * CHK * CHK;
#pragma unroll
  for (int i = 0; i < 8; ++i) {
    int t = mbase + i;
    float v = (s < t) ? LRC * (acc[i] + 1.0f) : 0.0f;
    Ab[(size_t)t * CHK + s] = v;
  }
}

// Sequential in-chunk triangular solve. Each thread owns one H-column; its
// diff history is private, held in LDS (C*H*4 = 192KB of the 320KB WGP LDS),
// so no barriers are needed. Also applies the b-state update.
__global__ void k_solve(const float* __restrict__ x, const float* __restrict__ P0,
                        const float* __restrict__ Amat, float* __restrict__ Pred,
                        float* __restrict__ Diff, float* __restrict__ bst, int chunk) {
  extern __shared__ float dsh[];   // [CHK][HH], column-private per thread
  int b = blockIdx.x;
  int h = threadIdx.x;             // 0..767
  const float* Xb = x + (size_t)b * SS * HH;
  const float* Ab = Amat + (size_t)b * CHK * CHK;
  float bsum = 0.0f;
  for (int t = 0; t < CHK; ++t) {
    int gt = chunk * CHK + t;
    float acc = P0[((size_t)b * CHK + t) * HH + h];
    const float* Arow = Ab + (size_t)t * CHK;
    for (int s = 0; s < t; ++s) acc -= Arow[s] * dsh[(size_t)s * HH + h];
    Pred[((size_t)b * CHK + t) * HH + h] = acc;          // pred_t
    float xn = (gt + 1 < SS) ? Xb[(size_t)(gt + 1) * HH + h] : 0.0f;
    float d = acc - xn;                                  // raw diff (recurrence)
    dsh[(size_t)t * HH + h] = d;
    float dl = (gt == SS - 1) ? 0.0f : d;                // lr-masked (state update)
    Diff[((size_t)b * CHK + t) * HH + h] = dl;
    bsum += dl;
  }
  bst[(size_t)b * HH + h] -= LRC * bsum;
}

// Wst -= LR * Diff^T * X_c   [H x H] per batch, K = C.
__global__ void k_wupdate(const float* __restrict__ x, const float* __restrict__ Diff,
                          float* __restrict__ Wst, int chunk) {
  int wid = (blockIdx.x * blockDim.x + threadIdx.x) >> 5;
  int lane = threadIdx.x & 31;
  int tn = wid % (HH / 16);
  int tm = (wid / (HH / 16)) % (HH / 16);
  int b  = wid / ((HH / 16) * (HH / 16));
  const float* Xc = x + ((size_t)b * SS + chunk * CHK) * HH;
  const float* Db = Diff + (size_t)b * CHK * HH;
  float* W = Wst + (size_t)b * HH * HH;
  v8f s = {};
  for (int k0 = 0; k0 < CHK; k0 += 4) {
    v2f a  = frag_kn(Db, HH, k0, tm * 16, lane);   // A[m,k] = Diff[k,m]
    v2f bb = frag_kn(Xc, HH, k0, tn * 16, lane);   // B[k,n] = Xc[k,n]
    s = wmma4(a, bb, s);
  }
  v8f acc = load_tile(W, HH, tm * 16, tn * 16, lane);
#pragma unroll
  for (int i = 0; i < 8; ++i) acc[i] -= LRC * s[i];
  store_tile(W, HH, tm * 16, tn * 16, acc, lane);
}

// Gated = sigmoid(Pred*gate_w^T + gate_b) * (Pred*up_w^T + up_b)  [C x 4H].
__global__ void k_mlp1(const float* __restrict__ Pred,
                       const float* __restrict__ gw, const float* __restrict__ gb,
                       const float* __restrict__ uw, const float* __restrict__ ub,
                       float* __restrict__ Gt) {
  int wid = (blockIdx.x * blockDim.x + threadIdx.x) >> 5;
  int lane = threadIdx.x & 31;
  int tn = wid % (H4 / 16);
  int tm = (wid / (H4 / 16)) % (CHK / 16);
  int b  = wid / ((H4 / 16) * (CHK / 16));
  const float* Pb = Pred + (size_t)b * CHK * HH;
  v8f g = {}, u = {};
  for (int k0 = 0; k0 < HH; k0 += 4) {
    v2f a  = frag_nk(Pb, HH, tm * 16, k0, lane);
    v2f bg = frag_nk(gw, HH, tn * 16, k0, lane);   // gate_w is [4H x H]
    v2f bu = frag_nk(uw, HH, tn * 16, k0, lane);
    g = wmma4(a, bg, g);
    u = wmma4(a, bu, u);
  }
  int n = tn * 16 + (lane & 15);
  float gbias = gb[n], ubias = ub[n];
  v8f r;
#pragma unroll
  for (int i = 0; i < 8; ++i) {
    float zg = g[i] + gbias;
    float zu = u[i] + ubias;
    r[i] = (1.0f / (1.0f + __expf(-zg))) * zu;
  }
  store_tile(Gt + (size_t)b * CHK * H4, H4, tm * 16, tn * 16, r, lane);
}

// Ypre = Gated*down_w^T + down_b + Pred   [C x H], K = 4H.
__global__ void k_mlp2(const float* __restrict__ Gt, const float* __restrict__ dw,
                       const float* __restrict__ db, const float* __restrict__ Pred,
                       float* __restrict__ Ypre) {
  int wid = (blockIdx.x * blockDim.x + threadIdx.x) >> 5;
  int lane = threadIdx.x & 31;
  int tn = wid % (HH / 16);
  int tm = (wid / (HH / 16)) % (CHK / 16);
  int b  = wid / ((HH / 16) * (CHK / 16));
  const float* Gb = Gt + (size_t)b * CHK * H4;
  v8f acc = {};
  for (int k0 = 0; k0 < H4; k0 += 4) {
    if ((k0 & 63) == 0) __builtin_prefetch(Gb + (size_t)(tm * 16) * H4 + k0 + 64, 0, 0);
    v2f a  = frag_nk(Gb, H4, tm * 16, k0, lane);
    v2f bb = frag_nk(dw, H4, tn * 16, k0, lane);   // down_w is [H x 4H]
    acc = wmma4(a, bb, acc);
  }
  int n = tn * 16 + (lane & 15);
  int mbase = tm * 16 + ((lane >> 4) << 3);
  float bias = db[n];
  const float* Pb = Pred + (size_t)b * CHK * HH;
  float* Yb = Ypre + (size_t)b * CHK * HH;
#pragma unroll
  for (int i = 0; i < 8; ++i) {
    int m = mbase + i;
    Yb[(size_t)m * HH + n] = acc[i] + bias + Pb[(size_t)m * HH + n];
  }
}

// Row LayerNorm of Ypre -> out[b, chunk*C + t, :].  One 256-thread block/row.
__global__ void k_ln(const float* __restrict__ Y, const float* __restrict__ nw,
                     const float* __restrict__ nb, float* __restrict__ out, int chunk) {
  __shared__ float s1[256], s2[256];
  int row = blockIdx.x;             // 0 .. B*CHK-1
  int b = row / CHK, t = row % CHK;
  const float* y = Y + ((size_t)b * CHK + t) * HH;
  int tid = threadIdx.x;
  float v0 = y[tid], v1 = y[tid + 256], v2 = y[tid + 512];
  s1[tid] = v0 + v1 + v2;
  s2[tid] = v0 * v0 + v1 * v1 + v2 * v2;
  __syncthreads();
  for (int off = 128; off > 0; off >>= 1) {
    if (tid < off) { s1[tid] += s1[tid + off]; s2[tid] += s2[tid + off]; }
    __syncthreads();
  }
  float mu  = s1[0] * (1.0f / HH);
  float var = s2[0] * (1.0f / HH) - mu * mu;
  float r = rsqrtf(var + EPSC);
  int gt = chunk * CHK + t;
  float* o = out + ((size_t)b * SS + gt) * HH;
  o[tid]       = (v0 - mu) * r * nw[tid]       + nb[tid];
  o[tid + 256] = (v1 - mu) * r * nw[tid + 256] + nb[tid + 256];
  o[tid + 512] = (v2 - mu) * r * nw[tid + 512] + nb[tid + 512];
}

// ---------------- launch ----------------
extern "C" void kernel_launch(void* const* d_in, const int* in_sizes, int n_in,
                              void* d_out, int out_size, void* d_ws, size_t ws_size,
                              hipStream_t stream) {
  const float* x      = (const float*)d_in[0];
  const float* W_init = (const float*)d_in[1];
  const float* b_init = (const float*)d_in[2];
  const float* gate_w = (const float*)d_in[3];
  const float* gate_b = (const float*)d_in[4];
  const float* up_w   = (const float*)d_in[5];
  const float* up_b   = (const float*)d_in[6];
  const float* down_w = (const float*)d_in[7];
  const float* down_b = (const float*)d_in[8];
  const float* norm_w = (const float*)d_in[9];
  const float* norm_b = (const float*)d_in[10];
  float* out = (float*)d_out;

  // Workspace layout (fp32): ~7.9 MB total.
  float* ws   = (float*)d_ws;
  float* Wst  = ws;  ws += (size_t)BB * HH * HH;   // fast-weight state
  float* bst  = ws;  ws += (size_t)BB * HH;        // fast-bias state
  float* P0   = ws;  ws += (size_t)BB * CHK * HH;
  float* Pred = ws;  ws += (size_t)BB * CHK * HH;
  float* Diff = ws;  ws += (size_t)BB * CHK * HH;
  float* Amat = ws;  ws += (size_t)BB * CHK * CHK;
  float* Gt   = ws;  ws += (size_t)BB * CHK * H4;
  float* Ypre = ws;  ws += (size_t)BB * CHK * HH;

  k_init<<<(HH * HH + 255) / 256, 256, 0, stream>>>(W_init, b_init, Wst, bst);

  const int NCH = SS / CHK;   // 4 sequential chunk iterations
  for (int c = 0; c < NCH; ++c) {
    // waves: B * (C/16) * (H/16) = 384 -> 48 blocks of 8 waves
    k_p0<<<48, 256, 0, stream>>>(x, Wst, bst, P0, c);
    // waves: B * 4 * 4 = 32 -> 4 blocks
    k_attn<<<4, 256, 0, stream>>>(x, Amat, c);
    // 1 block per batch, 768 threads, 192KB dynamic LDS
    k_solve<<<BB, HH, CHK * HH * sizeof(float), stream>>>(x, P0, Amat, Pred, Diff, bst, c);
    // waves: B * 48 * 48 = 4608 -> 576 blocks
    k_wupdate<<<576, 256, 0, stream>>>(x, Diff, Wst, c);
    // waves: B * 4 * 192 = 1536 -> 192 blocks
    k_mlp1<<<192, 256, 0, stream>>>(Pred, gate_w, gate_b, up_w, up_b, Gt);
    // waves: B * 4 * 48 = 384 -> 48 blocks
    k_mlp2<<<48, 256, 0, stream>>>(Gt, down_w, down_b, Pred, Ypre);
    // B*C = 128 rows
    k_ln<<<BB * CHK, 256, 0, stream>>>(Ypre, norm_w, norm_b, out, c);
  }
}